// ResidualVectorQuantizer_83846351553222
// MI455X (gfx1250) — compile-verified
//
#include <hip/hip_runtime.h>

typedef float v2f __attribute__((ext_vector_type(2)));
typedef float v8f __attribute__((ext_vector_type(8)));

#define N_CB        4
#define N_EMBED     1024
#define DIM         64
#define M_TOTAL     131072          // 8*32*32 patches * 16 px
#define Z_ELEMS     8388608         // 8*64*128*128
#define ROWS_PER_BLOCK 128          // 8 waves * 16 rows
#define CHUNK_CODES 64              // codes per LDS chunk (double buffered)
#define N_CHUNKS    (N_EMBED / CHUNK_CODES)          // 16 per stage
#define TOT_CHUNKS  (N_CB * N_CHUNKS)                // 64 overall
#define LDS_STRIDE  68              // 64 data + [64]=norm + 3 pad (conflict-free b64)

// ---------------- per-code squared norms ----------------
__global__ void rvq_norms_kernel(const float* __restrict__ cb,
                                 float* __restrict__ norms) {
    int i = blockIdx.x * blockDim.x + threadIdx.x;
    if (i >= N_CB * N_EMBED) return;
    const float* r = cb + i * DIM;
    float s = 0.f;
#pragma unroll
    for (int k = 0; k < DIM; ++k) s += r[k] * r[k];
    norms[i] = s;
}

// issue one chunk's async global->LDS copies: 4 x b128 (data) + 1 x b32 (norm)
// per thread => exactly 5 ASYNCcnt increments per wave (uniform).
__device__ __forceinline__ void issue_chunk(const float* cbS, const float* nrmS,
                                            unsigned ldsBuf, int chunk, int tid) {
#pragma unroll
    for (int i = 0; i < 4; ++i) {
        const int f   = i * 256 + tid;              // b128 unit in 64x64 chunk
        const int row = f >> 4;
        const int c4  = f & 15;
        unsigned gOff = (unsigned)((chunk * CHUNK_CODES * DIM + row * DIM + c4 * 4) * 4);
        unsigned lOff = ldsBuf + (unsigned)((row * LDS_STRIDE + c4 * 4) * 4);
        asm volatile("global_load_async_to_lds_b128 %0, %1, %2"
                     :: "v"(lOff), "v"(gOff), "s"(cbS) : "memory");
    }
    {   // per-code norm into the row padding slot [64]
        const int row = tid & (CHUNK_CODES - 1);    // duplicated writes, same value
        unsigned gOff = (unsigned)((chunk * CHUNK_CODES + row) * 4);
        unsigned lOff = ldsBuf + (unsigned)((row * LDS_STRIDE + DIM) * 4);
        asm volatile("global_load_async_to_lds_b32 %0, %1, %2"
                     :: "v"(lOff), "v"(gOff), "s"(nrmS) : "memory");
    }
}

// ---------------- main RVQ kernel ----------------
__global__ __launch_bounds__(256, 1)
void rvq_main_kernel(const float* __restrict__ z,
                     const float* __restrict__ cb,
                     const float* __restrict__ norms,
                     int*   __restrict__ counts,     // [4][1024]
                     float* __restrict__ sumsq,      // [4]
                     float* __restrict__ outZq,      // [8,64,128,128]
                     float* __restrict__ outIdx) {   // [M][4] as float
    __shared__ __align__(16) float ldsCB[2][CHUNK_CODES * LDS_STRIDE];
    __shared__ int sIdx[8][16];

    const int tid  = threadIdx.x;
    const int wv   = tid >> 5;
    const int lane = tid & 31;
    const int h    = lane >> 4;        // wave half (K split / M split)
    const int col  = lane & 15;        // N column / row-in-tile

    const unsigned ldsA0 = (unsigned)(uintptr_t)&ldsCB[0][0];
    const unsigned ldsA1 = (unsigned)(uintptr_t)&ldsCB[1][0];

    const int rowBase = blockIdx.x * ROWS_PER_BLOCK + wv * 16;
    const int m   = rowBase + col;     // this lane's vector row
    const int p   = m >> 4;
    const int q   = m & 15;
    const int b   = p >> 10;
    const int rem = p & 1023;
    const int hi  = rem >> 5;
    const int wi  = rem & 31;
    const int spatial = b * (64 * 16384) + (hi * 4 + (q >> 2)) * 128 + wi * 4 + (q & 3);

    v2f a[16], zq[16];
#pragma unroll
    for (int t = 0; t < 16; ++t) {
        const int c0 = 4 * t + 2 * h;
        a[t].x = z[spatial + c0 * 16384];
        a[t].y = z[spatial + (c0 + 1) * 16384];
        zq[t].x = 0.f; zq[t].y = 0.f;
    }

    float best[8]; int bidx[8];

    // ---- software pipeline over 64 chunks (4 stages x 16 chunks) ----
    issue_chunk(cb, norms, ldsA0, 0, tid);          // prologue: chunk 0 -> buf0

    for (int g = 0; g < TOT_CHUNKS; ++g) {
        const int stage = g >> 4;
        const int chunk = g & (N_CHUNKS - 1);
        const float* cbS = cb + stage * (N_EMBED * DIM);

        if (chunk == 0) {
#pragma unroll
            for (int r = 0; r < 8; ++r) { best[r] = 3.0e38f; bidx[r] = 0; }
        }

        if (g + 1 < TOT_CHUNKS) {       // issue next chunk into other buffer
            const int gn = g + 1;
            issue_chunk(cb + (gn >> 4) * (N_EMBED * DIM),
                        norms + (gn >> 4) * N_EMBED,
                        (gn & 1) ? ldsA1 : ldsA0, gn & (N_CHUNKS - 1), tid);
            asm volatile("s_wait_asynccnt 0x5" ::: "memory");
        } else {
            asm volatile("s_wait_asynccnt 0x0" ::: "memory");
        }
        __syncthreads();                 // current buffer valid block-wide

        const float* buf = &ldsCB[g & 1][0];
#pragma unroll
        for (int nt = 0; nt < CHUNK_CODES / 16; ++nt) {
            const int n = nt * 16 + col;
            v8f acc0 = {}, acc1 = {};
#pragma unroll
            for (int t = 0; t < 8; ++t) {
                v2f bf = *reinterpret_cast<const v2f*>(&buf[n * LDS_STRIDE + 4 * t + 2 * h]);
                acc0 = __builtin_amdgcn_wmma_f32_16x16x4_f32(
                    false, a[t], false, bf, (short)0, acc0, false, false);
            }
#pragma unroll
            for (int t = 8; t < 16; ++t) {
                v2f bf = *reinterpret_cast<const v2f*>(&buf[n * LDS_STRIDE + 4 * t + 2 * h]);
                acc1 = __builtin_amdgcn_wmma_f32_16x16x4_f32(
                    false, a[t], false, bf, (short)0, acc1, false, false);
            }
            const float nrm = buf[n * LDS_STRIDE + DIM];     // ||e||^2 from LDS
            const int codeBase = chunk * CHUNK_CODES + nt * 16 + col;
#pragma unroll
            for (int r = 0; r < 8; ++r) {
                float s = fmaf(-2.f, acc0[r], fmaf(-2.f, acc1[r], nrm));
                if (s < best[r]) { best[r] = s; bidx[r] = codeBase; }
            }
        }

        if (chunk == N_CHUNKS - 1) {
            // ---- stage epilogue: argmin across lanes, residual update ----
#pragma unroll
            for (int d = 1; d < 16; d <<= 1) {
#pragma unroll
                for (int r = 0; r < 8; ++r) {
                    float os = __shfl_xor(best[r], d, 32);
                    int   oi = __shfl_xor(bidx[r], d, 32);
                    if (os < best[r] || (os == best[r] && oi < bidx[r])) {
                        best[r] = os; bidx[r] = oi;
                    }
                }
            }
            if (col == 0) {
#pragma unroll
                for (int r = 0; r < 8; ++r) sIdx[wv][h * 8 + r] = bidx[r];
            }
            __syncthreads();
            const int myIdx = sIdx[wv][col];

            if (h == 0) {
                atomicAdd(&counts[stage * N_EMBED + myIdx], 1);
                outIdx[m * N_CB + stage] = (float)myIdx;
            }

            const float* e = cbS + myIdx * DIM;
            float ss = 0.f;
#pragma unroll
            for (int t = 0; t < 16; ++t) {
                const int c0 = 4 * t + 2 * h;
                v2f ev = *reinterpret_cast<const v2f*>(&e[c0]);
                a[t].x -= ev.x; a[t].y -= ev.y;
                zq[t].x += ev.x; zq[t].y += ev.y;
                ss = fmaf(a[t].x, a[t].x, ss);
                ss = fmaf(a[t].y, a[t].y, ss);
            }
#pragma unroll
            for (int d = 16; d > 0; d >>= 1) ss += __shfl_xor(ss, d, 32);
            if (lane == 0) atomicAdd(&sumsq[stage], ss);
        }
        __syncthreads();   // all waves done with buf[g&1] before its refill
    }

    // scatter z_q back into image layout
#pragma unroll
    for (int t = 0; t < 16; ++t) {
        const int c0 = 4 * t + 2 * h;
        outZq[spatial + c0 * 16384]       = zq[t].x;
        outZq[spatial + (c0 + 1) * 16384] = zq[t].y;
    }
}

// ---------------- loss finalize ----------------
__global__ void rvq_finalize_kernel(const int* __restrict__ counts,
                                    const float* __restrict__ sumsq,
                                    float* __restrict__ lossOut) {
    __shared__ float red[1024];
    const int t = threadIdx.x;
    float total = 0.f;
    for (int s = 0; s < N_CB; ++s) {
        float c = (float)counts[s * N_EMBED + t];
        float p = (c + 1e-5f) / ((float)M_TOTAL + 1e-5f * (float)N_EMBED);
        red[t] = -p * logf(p + 1e-5f);
        __syncthreads();
        for (int o = 512; o > 0; o >>= 1) {
            if (t < o) red[t] += red[t + o];
            __syncthreads();
        }
        if (t == 0) {
            float usage = logf((float)N_EMBED) - red[0];
            total += 0.01f * usage + 0.25f * (sumsq[s] / (float)Z_ELEMS);
        }
        __syncthreads();
    }
    if (t == 0) lossOut[0] = total;
}

extern "C" void kernel_launch(void* const* d_in, const int* in_sizes, int n_in,
                              void* d_out, int out_size, void* d_ws, size_t ws_size,
                              hipStream_t stream) {
    const float* z  = (const float*)d_in[0];   // [8,64,128,128]
    const float* cb = (const float*)d_in[1];   // [4,1024,64]

    float* out     = (float*)d_out;
    float* outZq   = out;                       // [8,64,128,128]
    float* outLoss = out + Z_ELEMS;             // scalar
    float* outIdx  = out + Z_ELEMS + 1;         // [M][4] as float

    // workspace: counts[4][1024] (int) | sumsq[4] | norms[4][1024]
    int*   counts = (int*)d_ws;
    float* sumsq  = (float*)d_ws + N_CB * N_EMBED;
    float* norms  = (float*)d_ws + N_CB * N_EMBED + N_CB;

    hipMemsetAsync(d_ws, 0, (N_CB * N_EMBED + N_CB) * sizeof(float), stream);

    rvq_norms_kernel<<<(N_CB * N_EMBED + 255) / 256, 256, 0, stream>>>(cb, norms);

    rvq_main_kernel<<<M_TOTAL / ROWS_PER_BLOCK, 256, 0, stream>>>(
        z, cb, norms, counts, sumsq, outZq, outIdx);

    rvq_finalize_kernel<<<1, 1024, 0, stream>>>(counts, sumsq, outLoss);
}